// EFGPND_48902497632301
// MI455X (gfx1250) — compile-verified
//
#include <hip/hip_runtime.h>

// ---------------------------------------------------------------------------
// GP posterior variance via batched complex CG with a dense Toeplitz operator.
//
// MI455X reasoning: the circulant embedding length 2M-1 = 8191 is prime, so
// the FFT path is Bluestein-on-GPU pain. Instead we materialize the Hermitian
// operator A^ = diag(ws) T diag(ws)/sigma^2 + I once as a dense M x M complex
// fp32 matrix (128 MB -> resident in the 192 MB global L2). Each CG iteration
// is then a (1024 x 4096)x(4096 x 4096) complex GEMM (4 real GEMMs,
// 1.37e11 FLOP/iter, 2.2e12 total) on V_WMMA_F32_16X16X4_F32 -- fp32 matrix
// cores, because CG to 1e-4 needs >13 mantissa bits (bf16 WMMA would cap
// accuracy near 1e-3).
//
// Data movement: G is stored k-pair-interleaved in global memory so that
// GLOBAL_LOAD_ASYNC_TO_LDS_B128 (ASYNCcnt) drops tiles into LDS already in
// WMMA B-fragment order (one ds_load_b64 per fragment, no packing movs).
// LDS is double-buffered: each chunk's copies are issued while the previous
// chunk's WMMAs run; one barrier per 16-wide K chunk.
// ---------------------------------------------------------------------------

#define MDIM      4096
#define BDIM      1024
#define NCIRC     8191          // 2*M - 1 (prime)
#define INV_SIGSQ 10.0f         // 1 / 0.1
#define CG_TOL2   1e-8f         // (1e-4)^2
#define NITER     16
#define TWO_PI_F  6.283185307179586f

typedef __attribute__((ext_vector_type(2))) float v2f;
typedef __attribute__((ext_vector_type(8))) float v8f;

// ---------------------------------------------------------------- c = ifft(v)
// Direct DFT, exact phase via integer (k*n mod N): c[n] = (1/N) sum v[k] e^{+2pi i kn/N}
__global__ void dft_ifft_kernel(const float* __restrict__ vhat,
                                float* __restrict__ c_re,
                                float* __restrict__ c_im) {
    int n = blockIdx.x * blockDim.x + threadIdx.x;
    if (n >= NCIRC) return;
    float sr = 0.f, si = 0.f;
    unsigned r = 0;                       // (k*n) mod N, exact
    for (int k = 0; k < NCIRC; ++k) {
        float ang = TWO_PI_F * ((float)r * (1.0f / (float)NCIRC));
        float s, c;
        __sincosf(ang, &s, &c);
        float v = vhat[k];
        sr += v * c;
        si += v * s;
        r += (unsigned)n;
        if (r >= NCIRC) r -= NCIRC;
    }
    c_re[n] = sr * (1.0f / (float)NCIRC);
    c_im[n] = si * (1.0f / (float)NCIRC);
}

// --------------------------------------- G (k-pair interleaved) = A^[n,k]
// A^[n,k] = ws[n]*ws[k]*c[(n-k) mod N]/sigma^2 + delta_{nk}.
// Storage: G'[(k>>1)*2M + n*2 + (k&1)] so a 16-byte run holds {2 cols x 2 k},
// exactly the WMMA B-fragment order after an async B128 copy into LDS.
__global__ void build_g_kernel(const float* __restrict__ ws,
                               const float* __restrict__ c_re,
                               const float* __restrict__ c_im,
                               float* __restrict__ g_re,
                               float* __restrict__ g_im) {
    const int total = MDIM * MDIM;
    for (int idx = blockIdx.x * blockDim.x + threadIdx.x; idx < total;
         idx += gridDim.x * blockDim.x) {
        int k = idx >> 12;
        int n = idx & (MDIM - 1);
        int d = n - k;
        if (d < 0) d += NCIRC;
        float w = ws[n] * ws[k] * INV_SIGSQ;
        size_t gi = (size_t)(k >> 1) * (2 * MDIM) + n * 2 + (k & 1);
        g_re[gi] = w * c_re[d] + ((n == k) ? 1.0f : 0.0f);
        g_im[gi] = w * c_im[d];
    }
}

// ------------------------------------- fx = exp(2 pi i x xi); r0 = p0 = ws*conj(fx)
__global__ void init_kernel(const float* __restrict__ x_new,
                            const float* __restrict__ xis,
                            const float* __restrict__ ws,
                            float* __restrict__ fx_re, float* __restrict__ fx_im,
                            float* __restrict__ r_re,  float* __restrict__ r_im,
                            float* __restrict__ p_re,  float* __restrict__ p_im,
                            float* __restrict__ x_re,  float* __restrict__ x_im) {
    const int total = BDIM * MDIM;
    for (int idx = blockIdx.x * blockDim.x + threadIdx.x; idx < total;
         idx += gridDim.x * blockDim.x) {
        int b = idx >> 12;
        int m = idx & (MDIM - 1);
        float ph = TWO_PI_F * x_new[b] * xis[m];
        float s, c;
        __sincosf(ph, &s, &c);
        fx_re[idx] = c;
        fx_im[idx] = s;
        float w  = ws[m];
        float rr = w * c;
        float ri = -w * s;
        r_re[idx] = rr;  r_im[idx] = ri;
        p_re[idx] = rr;  p_im[idx] = ri;
        x_re[idx] = 0.f; x_im[idx] = 0.f;
    }
}

// ------------------------------------------------- async staging primitives
__device__ __forceinline__ void stage_async(
    unsigned lAre, unsigned lAim,
    unsigned lBre0, unsigned lBre1, unsigned lBim0, unsigned lBim1,
    unsigned oA, unsigned oB0, unsigned oB1,
    const float* Pre, const float* Pim,
    const float* Gre, const float* Gim) {
    asm volatile(
        "global_load_async_to_lds_b128 %0, %6, %9\n"
        "global_load_async_to_lds_b128 %1, %6, %10\n"
        "global_load_async_to_lds_b128 %2, %7, %11\n"
        "global_load_async_to_lds_b128 %3, %8, %11\n"
        "global_load_async_to_lds_b128 %4, %7, %12\n"
        "global_load_async_to_lds_b128 %5, %8, %12\n"
        :
        : "v"(lAre), "v"(lAim), "v"(lBre0), "v"(lBre1), "v"(lBim0), "v"(lBim1),
          "v"(oA), "v"(oB0), "v"(oB1),
          "s"(Pre), "s"(Pim), "s"(Gre), "s"(Gim)
        : "memory");
}

__device__ __forceinline__ void wait_async0() {
    asm volatile("s_wait_asynccnt 0x0" ::: "memory");
}

// --------------------------------------------------------------- WMMA CGEMM
// Ap(B x M) = P(B x M) * G(M[k] x M[n]) complex, K = 4096.
// Block tile 64 x 128, 8 wave32 waves (2 x 4 grid of 16x64 wave tiles).
// Double-buffered LDS, async copies overlapped with compute, one barrier
// per 16-wide K chunk; 64x v_wmma_f32_16x16x4_f32 per chunk.
__global__ __launch_bounds__(256, 1)
void cgemm_kernel(const float* __restrict__ Pre, const float* __restrict__ Pim,
                  const float* __restrict__ Gre, const float* __restrict__ Gim,
                  float* __restrict__ Ore, float* __restrict__ Oim) {
    __shared__ float sAre[2][64 * 16];   // [buf][row][k]
    __shared__ float sAim[2][64 * 16];
    __shared__ float sBre[2][8 * 256];   // [buf][kpair][col*2 + (k&1)]
    __shared__ float sBim[2][8 * 256];

    const int tid     = threadIdx.x;
    const int lane    = tid & 31;
    const int wid     = tid >> 5;
    const int waveRow = wid & 3;     // 4 row groups of 16
    const int waveCol = wid >> 2;    // 2 col groups of 64
    const int half    = lane >> 4;   // lane half selects K pair / M half
    const int mrow    = lane & 15;

    const int rowBase = blockIdx.y * 64;
    const int colBase = blockIdx.x * 128;

    // ---- per-thread async staging coordinates
    const int ar = tid >> 2;                         // A row 0..63
    const int aq = tid & 3;                          // A float4 within chunk
    const int ktp0 = tid >> 6,          c0 = tid & 63;           // B part 0
    const int ktp1 = (tid + 256) >> 6,  c1 = (tid + 256) & 63;   // B part 1

    unsigned lAre[2], lAim[2], lBre0[2], lBre1[2], lBim0[2], lBim1[2];
#pragma unroll
    for (int b = 0; b < 2; ++b) {
        lAre[b]  = (unsigned)(size_t)&sAre[b][ar * 16 + aq * 4];
        lAim[b]  = (unsigned)(size_t)&sAim[b][ar * 16 + aq * 4];
        lBre0[b] = (unsigned)(size_t)&sBre[b][ktp0 * 256 + c0 * 4];
        lBre1[b] = (unsigned)(size_t)&sBre[b][ktp1 * 256 + c1 * 4];
        lBim0[b] = (unsigned)(size_t)&sBim[b][ktp0 * 256 + c0 * 4];
        lBim1[b] = (unsigned)(size_t)&sBim[b][ktp1 * 256 + c1 * 4];
    }

    // byte offsets into P / G' for this thread's chunk-0 copy
    unsigned gA  = (unsigned)(((rowBase + ar) * MDIM + aq * 4) * 4);
    unsigned gB0 = (unsigned)(ktp0 * (2 * MDIM * 4) + colBase * 8 + c0 * 16);
    unsigned gB1 = (unsigned)(ktp1 * (2 * MDIM * 4) + colBase * 8 + c1 * 16);
    const unsigned stepA = 16 * 4;            // +16 floats along a P row
    const unsigned stepB = 16 * MDIM * 4;     // +8 k-pair rows of G'

    v8f accRe[4] = {};
    v8f accIm[4] = {};

    // prologue: fill buffer 0 with chunk 0
    stage_async(lAre[0], lAim[0], lBre0[0], lBre1[0], lBim0[0], lBim1[0],
                gA, gB0, gB1, Pre, Pim, Gre, Gim);
    gA += stepA; gB0 += stepB; gB1 += stepB;

    for (int kk = 0; kk < MDIM; kk += 32) {
#pragma unroll
        for (int hb = 0; hb < 2; ++hb) {
            wait_async0();               // this wave's copies into buf hb done
            __syncthreads();             // all waves' copies done + prior reads done
            if (kk + 16 * (hb + 1) < MDIM) {   // overlap next chunk into buf hb^1
                stage_async(lAre[hb ^ 1], lAim[hb ^ 1],
                            lBre0[hb ^ 1], lBre1[hb ^ 1],
                            lBim0[hb ^ 1], lBim1[hb ^ 1],
                            gA, gB0, gB1, Pre, Pim, Gre, Gim);
                gA += stepA; gB0 += stepB; gB1 += stepB;
            }
            // ---- 4 WMMA K-steps of 4 on buffer hb
#pragma unroll
            for (int ks = 0; ks < 4; ++ks) {
                const int kf = ks * 4 + half * 2;   // frag K = kf + {0,1}
                v2f aRe  = *(const v2f*)&sAre[hb][(waveRow * 16 + mrow) * 16 + kf];
                v2f aIm  = *(const v2f*)&sAim[hb][(waveRow * 16 + mrow) * 16 + kf];
                v2f aImN = -aIm;         // f32 WMMA has no A-neg; negate frag
                const int bro = (ks * 2 + half) * 256;
#pragma unroll
                for (int t = 0; t < 4; ++t) {
                    const int col = waveCol * 64 + t * 16 + mrow;
                    v2f bRe = *(const v2f*)&sBre[hb][bro + col * 2];
                    v2f bIm = *(const v2f*)&sBim[hb][bro + col * 2];
                    // Re += Pre*Gre - Pim*Gim ; Im += Pre*Gim + Pim*Gre
                    accRe[t] = __builtin_amdgcn_wmma_f32_16x16x4_f32(
                        false, aRe,  false, bRe, (short)0, accRe[t], false, false);
                    accRe[t] = __builtin_amdgcn_wmma_f32_16x16x4_f32(
                        false, aImN, false, bIm, (short)0, accRe[t], false, false);
                    accIm[t] = __builtin_amdgcn_wmma_f32_16x16x4_f32(
                        false, aRe,  false, bIm, (short)0, accIm[t], false, false);
                    accIm[t] = __builtin_amdgcn_wmma_f32_16x16x4_f32(
                        false, aIm,  false, bRe, (short)0, accIm[t], false, false);
                }
            }
        }
    }

    // ---- write out: C layout — VGPR v, lane half h => row = h*8 + v, col = mrow
    const int outRow0 = rowBase + waveRow * 16 + half * 8;
#pragma unroll
    for (int t = 0; t < 4; ++t) {
        const int col = colBase + waveCol * 64 + t * 16 + mrow;
#pragma unroll
        for (int v = 0; v < 8; ++v) {
            Ore[(size_t)(outRow0 + v) * MDIM + col] = accRe[t][v];
            Oim[(size_t)(outRow0 + v) * MDIM + col] = accIm[t][v];
        }
    }
}

// -------------------------------------------------------- per-row reductions
__global__ void reduce_rs_kernel(const float* __restrict__ r_re,
                                 const float* __restrict__ r_im,
                                 float* __restrict__ rs_out,
                                 float* __restrict__ tol2_out) {
    __shared__ float sbuf[256];
    const int b = blockIdx.x, t = threadIdx.x;
    const float* pr = r_re + (size_t)b * MDIM;
    const float* pi = r_im + (size_t)b * MDIM;
    float s = 0.f;
    for (int m = t; m < MDIM; m += 256) {
        float a = pr[m], c = pi[m];
        s += a * a + c * c;
    }
    sbuf[t] = s;
    __syncthreads();
    for (int off = 128; off > 0; off >>= 1) {
        if (t < off) sbuf[t] += sbuf[t + off];
        __syncthreads();
    }
    if (t == 0) {
        rs_out[b] = sbuf[0];
        if (tol2_out) tol2_out[b] = CG_TOL2 * sbuf[0];
    }
}

__global__ void reduce_denom_kernel(const float* __restrict__ p_re,
                                    const float* __restrict__ p_im,
                                    const float* __restrict__ a_re,
                                    const float* __restrict__ a_im,
                                    float* __restrict__ denom) {
    __shared__ float sbuf[256];
    const int b = blockIdx.x, t = threadIdx.x;
    const size_t base = (size_t)b * MDIM;
    float s = 0.f;
    for (int m = t; m < MDIM; m += 256)
        s += p_re[base + m] * a_re[base + m] + p_im[base + m] * a_im[base + m];
    sbuf[t] = s;
    __syncthreads();
    for (int off = 128; off > 0; off >>= 1) {
        if (t < off) sbuf[t] += sbuf[t + off];
        __syncthreads();
    }
    if (t == 0) denom[b] = sbuf[0];
}

// ------------------------------------------------------------- CG scalar ops
__global__ void scalars_alpha_kernel(const float* __restrict__ rs,
                                     const float* __restrict__ tol2,
                                     const float* __restrict__ denom,
                                     float* __restrict__ alpha) {
    int b = blockIdx.x * blockDim.x + threadIdx.x;
    if (b < BDIM) {
        float r = rs[b];
        bool act = r > tol2[b];
        alpha[b] = act ? r / fmaxf(denom[b], 1e-30f) : 0.0f;
    }
}

__global__ void scalars_beta_kernel(float* __restrict__ rs,
                                    const float* __restrict__ tol2,
                                    const float* __restrict__ rsnew,
                                    float* __restrict__ beta,
                                    float* __restrict__ actf) {
    int b = blockIdx.x * blockDim.x + threadIdx.x;
    if (b < BDIM) {
        float r = rs[b];
        bool act = r > tol2[b];
        float rn = rsnew[b];
        beta[b] = act ? rn / fmaxf(r, 1e-30f) : 0.0f;
        actf[b] = act ? 1.0f : 0.0f;
        rs[b]   = act ? rn : r;
    }
}

// --------------------------------------------------------- elementwise axpys
__global__ void update_xr_kernel(const float* __restrict__ alpha,
                                 const float* __restrict__ p_re, const float* __restrict__ p_im,
                                 const float* __restrict__ a_re, const float* __restrict__ a_im,
                                 float* __restrict__ x_re, float* __restrict__ x_im,
                                 float* __restrict__ r_re, float* __restrict__ r_im) {
    const int total = BDIM * MDIM;
    for (int idx = blockIdx.x * blockDim.x + threadIdx.x; idx < total;
         idx += gridDim.x * blockDim.x) {
        float al = alpha[idx >> 12];
        x_re[idx] += al * p_re[idx];
        x_im[idx] += al * p_im[idx];
        r_re[idx] -= al * a_re[idx];
        r_im[idx] -= al * a_im[idx];
    }
}

__global__ void update_p_kernel(const float* __restrict__ beta,
                                const float* __restrict__ actf,
                                const float* __restrict__ r_re, const float* __restrict__ r_im,
                                float* __restrict__ p_re, float* __restrict__ p_im) {
    const int total = BDIM * MDIM;
    for (int idx = blockIdx.x * blockDim.x + threadIdx.x; idx < total;
         idx += gridDim.x * blockDim.x) {
        int b = idx >> 12;
        float be = beta[b];
        if (actf[b] != 0.0f) {
            p_re[idx] = r_re[idx] + be * p_re[idx];
            p_im[idx] = r_im[idx] + be * p_im[idx];
        }
    }
}

// ----------------------------------------- s2[b] = max(0, Re sum fx*ws*gamma)
__global__ void final_s2_kernel(const float* __restrict__ ws,
                                const float* __restrict__ fx_re,
                                const float* __restrict__ fx_im,
                                const float* __restrict__ x_re,
                                const float* __restrict__ x_im,
                                float* __restrict__ out) {
    __shared__ float sbuf[256];
    const int b = blockIdx.x, t = threadIdx.x;
    const size_t base = (size_t)b * MDIM;
    float s = 0.f;
    for (int m = t; m < MDIM; m += 256)
        s += ws[m] * (fx_re[base + m] * x_re[base + m] -
                      fx_im[base + m] * x_im[base + m]);
    sbuf[t] = s;
    __syncthreads();
    for (int off = 128; off > 0; off >>= 1) {
        if (t < off) sbuf[t] += sbuf[t + off];
        __syncthreads();
    }
    if (t == 0) out[b] = fmaxf(sbuf[0], 0.0f);
}

// ---------------------------------------------------------------------------
extern "C" void kernel_launch(void* const* d_in, const int* in_sizes, int n_in,
                              void* d_out, int out_size, void* d_ws, size_t ws_size,
                              hipStream_t stream) {
    const float* x_new = (const float*)d_in[0];   // (B,1)
    const float* xis   = (const float*)d_in[1];   // (M,1)
    const float* ws    = (const float*)d_in[2];   // (M,)
    const float* vhat  = (const float*)d_in[3];   // (2M-1,)
    float* out = (float*)d_out;

    float* base = (float*)d_ws;
    const size_t MM = (size_t)MDIM * MDIM;   // 16,777,216
    const size_t BM = (size_t)BDIM * MDIM;   //  4,194,304
    size_t off = 0;
    float* c_re  = base + off; off += 8192;
    float* c_im  = base + off; off += 8192;
    float* g_re  = base + off; off += MM;
    float* g_im  = base + off; off += MM;
    float* fx_re = base + off; off += BM;
    float* fx_im = base + off; off += BM;
    float* x_re  = base + off; off += BM;
    float* x_im  = base + off; off += BM;
    float* r_re  = base + off; off += BM;
    float* r_im  = base + off; off += BM;
    float* p_re  = base + off; off += BM;
    float* p_im  = base + off; off += BM;
    float* a_re  = base + off; off += BM;
    float* a_im  = base + off; off += BM;
    float* rs    = base + off; off += BDIM;
    float* tol2  = base + off; off += BDIM;
    float* denom = base + off; off += BDIM;
    float* rsnew = base + off; off += BDIM;
    float* alpha = base + off; off += BDIM;
    float* beta  = base + off; off += BDIM;
    float* actf  = base + off; off += BDIM;

    // setup
    dft_ifft_kernel<<<(NCIRC + 255) / 256, 256, 0, stream>>>(vhat, c_re, c_im);
    build_g_kernel<<<8192, 256, 0, stream>>>(ws, c_re, c_im, g_re, g_im);
    init_kernel<<<4096, 256, 0, stream>>>(x_new, xis, ws, fx_re, fx_im,
                                          r_re, r_im, p_re, p_im, x_re, x_im);
    reduce_rs_kernel<<<BDIM, 256, 0, stream>>>(r_re, r_im, rs, tol2);

    // CG iterations
    dim3 ggrid(MDIM / 128, BDIM / 64);   // 32 x 16 workgroups
    for (int it = 0; it < NITER; ++it) {
        cgemm_kernel<<<ggrid, 256, 0, stream>>>(p_re, p_im, g_re, g_im, a_re, a_im);
        reduce_denom_kernel<<<BDIM, 256, 0, stream>>>(p_re, p_im, a_re, a_im, denom);
        scalars_alpha_kernel<<<4, 256, 0, stream>>>(rs, tol2, denom, alpha);
        update_xr_kernel<<<4096, 256, 0, stream>>>(alpha, p_re, p_im, a_re, a_im,
                                                   x_re, x_im, r_re, r_im);
        reduce_rs_kernel<<<BDIM, 256, 0, stream>>>(r_re, r_im, rsnew, nullptr);
        scalars_beta_kernel<<<4, 256, 0, stream>>>(rs, tol2, rsnew, beta, actf);
        update_p_kernel<<<4096, 256, 0, stream>>>(beta, actf, r_re, r_im, p_re, p_im);
    }

    final_s2_kernel<<<BDIM, 256, 0, stream>>>(ws, fx_re, fx_im, x_re, x_im, out);
}